// ContrastiveAlignmentLoss_26594437497388
// MI455X (gfx1250) — compile-verified
//
#include <hip/hip_runtime.h>

typedef __attribute__((ext_vector_type(2))) float v2f;
typedef __attribute__((ext_vector_type(4))) float v4f;
typedef __attribute__((ext_vector_type(8))) float v8f;

#define NB        256              // batch
#define N_ELEM    (512 * 512)      // elements per graph
#define QUARTERS  4
#define CHUNK     (N_ELEM / QUARTERS)   // 65536 floats per block
#define VEC4S     (CHUNK / 4)           // 16384 float4 loads per block
#define NUM_PAIRS 32640.0f              // 256*255/2

// ---------------- Phase 1: bandwidth-bound mean pooling ----------------
// grid = (256 batches, 4 chunks), block = 256 threads (8 wave32s).
// 268 MB read once at 23.3 TB/s ~= 11.5 us floor; b128 non-temporal loads
// (read-once data, larger than the 192 MB L2). Deterministic partials to d_ws.
__global__ __launch_bounds__(256) void pool_partial_kernel(
    const float* __restrict__ graph, float* __restrict__ partials) {
  const int b = blockIdx.x;
  const int q = blockIdx.y;
  const v4f* __restrict__ src =
      (const v4f*)(graph + (size_t)b * N_ELEM + (size_t)q * CHUNK);

  float s = 0.0f;
#pragma unroll 4
  for (int k = threadIdx.x; k < VEC4S; k += 256) {
    v4f v = __builtin_nontemporal_load(&src[k]);
    s += (v.x + v.y) + (v.z + v.w);
  }

  __shared__ float red[256];
  red[threadIdx.x] = s;
  __syncthreads();
#pragma unroll
  for (int off = 128; off > 0; off >>= 1) {
    if (threadIdx.x < off) red[threadIdx.x] += red[threadIdx.x + off];
    __syncthreads();
  }
  if (threadIdx.x == 0) partials[b * QUARTERS + q] = red[0];
}

// ---------------- Phase 2: pairwise hinge loss via WMMA outer product ----
// Single 256-thread block (8 waves). sim tile = A(16x4) x B(4x16) with only
// K=0 populated -> exact f32 rank-1 outer product gs_i * gs_j on the matrix
// unit. 136 upper-triangular tiles, exactly 17 per wave (uniform trip count
// => EXEC all 1s at every v_wmma). Inner epilogue is fully branchless:
// unconditional LDS label loads + cndmask selects for (same) and (i<j).
__global__ __launch_bounds__(256) void pair_loss_kernel(
    const float* __restrict__ partials, const long long* __restrict__ labels,
    float* __restrict__ out) {
  __shared__ float gs[NB];
  __shared__ int   lab[NB];
  __shared__ float red[256];

  const int t = threadIdx.x;
  gs[t] = (partials[t * 4 + 0] + partials[t * 4 + 1] +
           partials[t * 4 + 2] + partials[t * 4 + 3]) * (1.0f / (float)N_ELEM);
  lab[t] = (int)labels[t];
  __syncthreads();

  const int lane    = t & 31;
  const int wave    = t >> 5;
  const int halfsel = lane >> 4;   // 0: M=v, 1: M=v+8  (D-matrix layout)
  const int nidx    = lane & 15;   // N coordinate within tile
  float acc = 0.0f;

  for (int idx = wave; idx < 136; idx += 8) {   // uniform 17 iters per wave
    // decode linear upper-tri tile index -> (rt, ct), rt <= ct
    int rem = idx, rt = 0;
    while (rem >= 16 - rt) { rem -= 16 - rt; ++rt; }
    const int ct = rt + rem;

    // A: 16x4, lanes 0-15 hold M=lane {K=0,K=1}; lanes 16-31 {K=2,K=3}.
    // Only K=0 carries data -> exact rank-1 product.
    v2f a, bb;
    a.x  = (lane < 16) ? gs[rt * 16 + lane] : 0.0f;  a.y  = 0.0f;
    bb.x = (lane < 16) ? gs[ct * 16 + lane] : 0.0f;  bb.y = 0.0f;

    v8f c = {};
    c = __builtin_amdgcn_wmma_f32_16x16x4_f32(
        /*neg_a=*/false, a, /*neg_b=*/false, bb,
        /*c_mod=*/(short)0, c, /*reuse_a=*/false, /*reuse_b=*/false);

    const int j     = ct * 16 + nidx;
    const int labj  = lab[j];          // loop-invariant per tile: 1 ds_load
    const int ibase = rt * 16 + halfsel * 8;
#pragma unroll
    for (int v = 0; v < 8; ++v) {
      const int   i   = ibase + v;
      const float sim = c[v];
      const float pos = fmaxf(0.5f - sim, 0.0f);       // MARGIN - sim
      const float neg = fmaxf(sim - 0.5f, 0.0f);       // sim - (1 - MARGIN)
      const float contrib = (lab[i] == labj) ? pos : neg;  // weights = 1.0
      acc += (i < j) ? contrib : 0.0f;                 // branchless triu mask
    }
  }

  red[t] = acc;
  __syncthreads();
#pragma unroll
  for (int off = 128; off > 0; off >>= 1) {
    if (t < off) red[t] += red[t + off];
    __syncthreads();
  }
  if (t == 0) out[0] = red[0] * (1.0f / NUM_PAIRS);
}

extern "C" void kernel_launch(void* const* d_in, const int* in_sizes, int n_in,
                              void* d_out, int out_size, void* d_ws, size_t ws_size,
                              hipStream_t stream) {
  const float*     graph  = (const float*)d_in[0];       // [256, 512, 512] f32
  const long long* labels = (const long long*)d_in[1];   // [256] int64
  float*           out    = (float*)d_out;               // scalar f32
  float*           part   = (float*)d_ws;                // 1024 partial sums

  dim3 grid1(NB, QUARTERS);
  pool_partial_kernel<<<grid1, 256, 0, stream>>>(graph, part);
  pair_loss_kernel<<<1, 256, 0, stream>>>(part, labels, out);
}